// GCN3PoolNorm_74543452389558
// MI455X (gfx1250) — compile-verified
//
#include <hip/hip_runtime.h>
#include <hip/hip_bf16.h>

#define NN 50000
#define NE 1200000
#define FIN 128
#define HID 64
#define NCL 10
#define NG 256
#define EPSV 1e-5f

typedef float v2f __attribute__((ext_vector_type(2)));
typedef float v8f __attribute__((ext_vector_type(8)));

// ---------------- degree / dinv ----------------
__global__ void k_deg(const int* __restrict__ dst, float* __restrict__ deg, int E) {
    int e = blockIdx.x * blockDim.x + threadIdx.x;
    if (e < E) atomicAdd(&deg[dst[e]], 1.0f);
}

__global__ void k_dinv(float* __restrict__ deg, int N) {
    int n = blockIdx.x * blockDim.x + threadIdx.x;
    if (n < N) deg[n] = rsqrtf(deg[n] + 1.0f);
}

// ---------------- fp32 WMMA GEMM: C[M x 64] = A[M x K] @ W[K x 64] ----------------
// blockDim = 128 (4 waves). Wave w computes the 16x16 tile at (blockIdx.x*16, w*16).
// Requires M % 16 == 0, K % 4 == 0. Uses V_WMMA_F32_16X16X4_F32.
// A-frag layout: lanes 0-15 -> rows lm, K = {k, k+1}; lanes 16-31 -> rows lm, K = {k+2, k+3}.
// B-frag layout: lanes 0-15 -> cols lm, K = {k, k+1}; lanes 16-31 -> cols lm, K = {k+2, k+3}.
// D layout: VGPR r, lanes 0-15 -> (row0 + r, col0 + lm); lanes 16-31 -> (row0 + 8 + r, col0 + lm).
__global__ void k_gemm_wmma64(const float* __restrict__ A, const float* __restrict__ W,
                              float* __restrict__ C, int M, int K) {
    const int N = 64;
    int lane = threadIdx.x & 31;
    int wave = threadIdx.x >> 5;
    int half = lane >> 4;
    int lm   = lane & 15;
    int row0 = blockIdx.x * 16;
    int col0 = wave * 16;

    v8f acc = {0.f, 0.f, 0.f, 0.f, 0.f, 0.f, 0.f, 0.f};
    const float* ap = A + (size_t)(row0 + lm) * K + half * 2;
    const float* bp = W + (size_t)(half * 2) * N + col0 + lm;

    for (int k = 0; k < K; k += 4) {
        v2f a; a.x = ap[0]; a.y = ap[1];
        v2f b; b.x = bp[0]; b.y = bp[N];
        acc = __builtin_amdgcn_wmma_f32_16x16x4_f32(
            /*neg_a=*/false, a, /*neg_b=*/false, b,
            /*c_mod=*/(short)0, acc, /*reuse_a=*/false, /*reuse_b=*/false);
        ap += 4;
        bp += 4 * N;
    }

    float* cp = C + (size_t)(row0 + half * 8) * N + col0 + lm;
#pragma unroll
    for (int r = 0; r < 8; ++r) cp[(size_t)r * N] = acc[r];
}

// ---------------- edge scatter-add: agg[dst] += h[src] * dinv[src]*dinv[dst] ----------------
__global__ void k_edge_agg(const int* __restrict__ src, const int* __restrict__ dst,
                           const float* __restrict__ dinv, const float* __restrict__ h,
                           float* __restrict__ agg, long total) {
    long idx = (long)blockIdx.x * blockDim.x + threadIdx.x;
    if (idx >= total) return;
    int e = (int)(idx >> 6);
    int c = (int)(idx & 63);
    int s = src[e], d = dst[e];
    float coef = dinv[s] * dinv[d];
    atomicAdd(&agg[(size_t)d * HID + c], h[(size_t)s * HID + c] * coef);
}

// ---------------- fused self-loop + bias + BN statistics ----------------
// pre = agg + dinv[n]^2 * hW + b  (written back into agg); block-reduced sum/sumsq -> bns[0..127]
__global__ void k_combine_bn(float* __restrict__ agg, const float* __restrict__ hW,
                             const float* __restrict__ dinv, const float* __restrict__ bias,
                             float* __restrict__ bns, int N) {
    __shared__ float ssum[HID];
    __shared__ float ssq[HID];
    int t = threadIdx.x;
    if (t < HID) { ssum[t] = 0.f; ssq[t] = 0.f; }
    __syncthreads();

    long idx = (long)blockIdx.x * blockDim.x + t;
    int n = (int)(idx >> 6);
    int c = (int)(idx & 63);
    float v = 0.f;
    if (n < N) {
        float dv = dinv[n];
        v = agg[idx] + dv * dv * hW[idx] + bias[c];
        agg[idx] = v;
    }
    atomicAdd(&ssum[c], v);
    atomicAdd(&ssq[c], v * v);
    __syncthreads();
    if (t < HID) {
        atomicAdd(&bns[t], ssum[t]);
        atomicAdd(&bns[HID + t], ssq[t]);
    }
}

// bns: [sum(64), sumsq(64)] -> writes scale at bns[128..191], shift at bns[192..255]
__global__ void k_bn_finalize(float* __restrict__ bns, const float* __restrict__ gamma,
                              const float* __restrict__ beta, int N) {
    int c = threadIdx.x;
    if (c >= HID) return;
    float inv_n = 1.0f / (float)N;
    float mu  = bns[c] * inv_n;
    float ex2 = bns[HID + c] * inv_n;
    float var = ex2 - mu * mu;
    float rstd = rsqrtf(var + EPSV);
    float scale = gamma[c] * rstd;
    bns[128 + c] = scale;
    bns[192 + c] = beta[c] - mu * scale;
}

__global__ void k_apply_relu(const float* __restrict__ pre, const float* __restrict__ bns,
                             float* __restrict__ hout, long total) {
    long idx = (long)blockIdx.x * blockDim.x + threadIdx.x;
    if (idx >= total) return;
    int c = (int)(idx & 63);
    float v = pre[idx] * bns[128 + c] + bns[192 + c];
    hout[idx] = fmaxf(v, 0.f);
}

// ---------------- pooling ----------------
// h >= 0 (post-ReLU), so float max == integer max on bit patterns with 0-init,
// which also reproduces the reference's where(isfinite, pmax, 0) for empty graphs.
__global__ void k_pool(const float* __restrict__ h, const int* __restrict__ batch,
                       float* __restrict__ pmax, float* __restrict__ psum, long total) {
    long idx = (long)blockIdx.x * blockDim.x + threadIdx.x;
    if (idx >= total) return;
    int n = (int)(idx >> 6);
    int c = (int)(idx & 63);
    int g = batch[n];
    float v = h[idx];
    atomicMax((int*)&pmax[(size_t)g * HID + c], __float_as_int(v));
    atomicAdd(&psum[(size_t)g * HID + c], v);
}

__global__ void k_count(const int* __restrict__ batch, float* __restrict__ cnt, int N) {
    int n = blockIdx.x * blockDim.x + threadIdx.x;
    if (n < N) atomicAdd(&cnt[batch[n]], 1.0f);
}

__global__ void k_emb(const float* __restrict__ pmax, const float* __restrict__ psum,
                      const float* __restrict__ cnt, float* __restrict__ emb) {
    int idx = blockIdx.x * blockDim.x + threadIdx.x; // NG * HID
    if (idx >= NG * HID) return;
    int g = idx >> 6;
    int c = idx & 63;
    float s = psum[idx];
    float m = fmaxf(cnt[g], 1.0f);
    emb[(size_t)g * (3 * HID) + c]           = pmax[idx];
    emb[(size_t)g * (3 * HID) + HID + c]     = s / m;
    emb[(size_t)g * (3 * HID) + 2 * HID + c] = s;
}

// ---------------- MLP tail ----------------
__global__ void k_bias_relu(float* __restrict__ mh, const float* __restrict__ lb1) {
    int idx = blockIdx.x * blockDim.x + threadIdx.x; // NG * HID
    if (idx >= NG * HID) return;
    int c = idx & 63;
    mh[idx] = fmaxf(mh[idx] + lb1[c], 0.f);
}

__global__ void k_mlp2(const float* __restrict__ mh, const float* __restrict__ lw2,
                       const float* __restrict__ lb2, float* __restrict__ out) {
    int idx = blockIdx.x * blockDim.x + threadIdx.x; // NG * NCL
    if (idx >= NG * NCL) return;
    int g = idx / NCL;
    int k = idx % NCL;
    float s = lb2[k];
#pragma unroll 8
    for (int c = 0; c < HID; ++c) s += mh[(size_t)g * HID + c] * lw2[(size_t)c * NCL + k];
    out[idx] = s;
}

extern "C" void kernel_launch(void* const* d_in, const int* in_sizes, int n_in,
                              void* d_out, int out_size, void* d_ws, size_t ws_size,
                              hipStream_t stream) {
    const float* x     = (const float*)d_in[0];
    const int*   ei    = (const int*)d_in[1];
    const int*   batch = (const int*)d_in[2];
    const float* W1    = (const float*)d_in[3];
    const float* b1    = (const float*)d_in[4];
    const float* W2    = (const float*)d_in[5];
    const float* b2    = (const float*)d_in[6];
    const float* gamma = (const float*)d_in[7];
    const float* beta  = (const float*)d_in[8];
    const float* lw1   = (const float*)d_in[9];
    const float* lb1   = (const float*)d_in[10];
    const float* lw2   = (const float*)d_in[11];
    const float* lb2   = (const float*)d_in[12];
    const int* esrc = ei;
    const int* edst = ei + NE;
    float* out = (float*)d_out;

    // workspace layout (floats); total ~9.75M floats (~39 MB)
    float* ws   = (float*)d_ws;
    float* dinv = ws;                       // 50048 (deg accumulated here, then rsqrt'd in place)
    float* hbuf = dinv + 50048;             // 3,200,000  current activations
    float* tmp  = hbuf + (size_t)NN * HID;  // 3,200,000  h @ W
    float* agg  = tmp  + (size_t)NN * HID;  // 3,200,000  edge aggregate / pre-BN
    float* bns  = agg  + (size_t)NN * HID;  // 256: sum, sumsq, scale, shift
    float* pmax = bns + 256;                // 16384
    float* psum = pmax + NG * HID;          // 16384
    float* cnt  = psum + NG * HID;          // 256
    float* emb  = cnt + 256;                // 49152
    float* mh   = emb + NG * 3 * HID;       // 16384

    const long nh = (long)NN * HID;               // 3.2M
    const long ne_tot = (long)NE * HID;           // 76.8M
    const int  B = 256;
    const int  g_nh  = (int)((nh + B - 1) / B);   // 12500
    const int  g_ne  = (int)((ne_tot + B - 1) / B);

    // ---- degree / dinv ----
    hipMemsetAsync(dinv, 0, (size_t)NN * sizeof(float), stream);
    k_deg<<<(NE + B - 1) / B, B, 0, stream>>>(edst, dinv, NE);
    k_dinv<<<(NN + B - 1) / B, B, 0, stream>>>(dinv, NN);

    // ---- 3 GCN layers ----
    for (int layer = 0; layer < 3; ++layer) {
        const float* hin  = (layer == 0) ? x : hbuf;
        const int    K    = (layer == 0) ? FIN : HID;
        const float* Wm   = (layer == 0) ? W1 : W2;
        const float* bias = (layer == 0) ? b1 : b2;

        k_gemm_wmma64<<<NN / 16, 128, 0, stream>>>(hin, Wm, tmp, NN, K);

        hipMemsetAsync(agg, 0, (size_t)nh * sizeof(float), stream);
        k_edge_agg<<<g_ne, B, 0, stream>>>(esrc, edst, dinv, tmp, agg, ne_tot);

        hipMemsetAsync(bns, 0, 256 * sizeof(float), stream);
        k_combine_bn<<<g_nh, B, 0, stream>>>(agg, tmp, dinv, bias, bns, NN);
        k_bn_finalize<<<1, 64, 0, stream>>>(bns, gamma, beta, NN);
        k_apply_relu<<<g_nh, B, 0, stream>>>(agg, bns, hbuf, nh);
    }

    // ---- pooling ----
    hipMemsetAsync(pmax, 0, (size_t)NG * HID * sizeof(float), stream);
    hipMemsetAsync(psum, 0, (size_t)NG * HID * sizeof(float), stream);
    hipMemsetAsync(cnt, 0, NG * sizeof(float), stream);
    k_pool<<<g_nh, B, 0, stream>>>(hbuf, batch, pmax, psum, nh);
    k_count<<<(NN + B - 1) / B, B, 0, stream>>>(batch, cnt, NN);
    k_emb<<<(NG * HID + B - 1) / B, B, 0, stream>>>(pmax, psum, cnt, emb);

    // ---- MLP head ----
    k_gemm_wmma64<<<NG / 16, 128, 0, stream>>>(emb, lw1, mh, NG, 3 * HID);
    k_bias_relu<<<(NG * HID + B - 1) / B, B, 0, stream>>>(mh, lb1);
    k_mlp2<<<(NG * NCL + B - 1) / B, B, 0, stream>>>(mh, lw2, lb2, out);
}